// BlockSharedRounding_635655160009
// MI455X (gfx1250) — compile-verified
//
#include <hip/hip_runtime.h>
#include <stdint.h>

// ---------------------------------------------------------------------------
// FP4 (E2M1) block soft-STE quantizer for MI455X (gfx1250).
//
// Memory-bound: 134 MB read + 268 MB written + 4 MB delta ~ 406 MB of traffic,
// ~17 us floor at 23.3 TB/s HBM vs ~1 GFLOP of VALU work (~20x memory bound).
// Strategy: per-wave double-buffered CDNA5 async global->LDS copies
// (ASYNCcnt / s_wait_asynccnt), ds_load_b128 consumption, 128-bit NT stores.
// No WMMA: no matmul structure; matrix ops cannot help a streaming kernel.
// ---------------------------------------------------------------------------

typedef float v4f __attribute__((ext_vector_type(4)));
typedef int   v4i __attribute__((ext_vector_type(4)));

#define CHUNK_V4 32u                 // wave-tile: 32 float4 = 128 elems = 4 quant blocks
#define WAVES_PB 8u                  // 256 threads / wave32
#define NTILES   (WAVES_PB * 2u)     // double buffer per wave

// Generic pointer -> 32-bit LDS byte offset (addrspacecast to AS3 + ptrtoint).
// This is what the async-to-LDS instructions take in VDST.
typedef __attribute__((address_space(3))) const char* lds_cptr_t;
__device__ __forceinline__ uint32_t lds_offset(const void* p)
{
    return (uint32_t)(uintptr_t)(lds_cptr_t)p;
}

// E2M1 decision bounds 0.25,0.75,1.25,1.75,2.5,3.5,5.0; values 0,.5,1,1.5,2,3,4,6.
__device__ __forceinline__ void quant1(float xin, float delta,
                                       float& outv, int& outk)
{
    float a = fabsf(xin) + delta;
    int k = (a > 0.25f) + (a > 0.75f) + (a > 1.25f) + (a > 1.75f)
          + (a > 2.5f)  + (a > 3.5f)  + (a > 5.0f);          // ordinal 0..7
    // VALUES[k] (hard / v_up):
    float up = (k <= 4) ? 0.5f * (float)k
                        : ((k == 5) ? 3.0f : ((k == 6) ? 4.0f : 6.0f));
    // v_dn = VALUES[k-1] for interior bins (k=0/7 overridden below)
    float dn = up - ((k <= 4) ? 0.5f : 1.0f);
    // bin midpoint c_k = 0.5*(bounds[k-1]+bounds[k]) for k=1..6
    float ck = (k <= 1) ? 0.5f   : (k == 2) ? 1.0f  : (k == 3) ? 1.5f
             : (k == 4) ? 2.125f : (k == 5) ? 3.0f  : 4.25f;
    // sigmoid(10*(a - c_k))
    float p    = __builtin_amdgcn_rcpf(1.0f + __expf(-10.0f * (a - ck)));
    float soft = fmaf(p, up - dn, dn);
    float mix  = soft + (up - soft);          // soft-STE forward value (== hard)
    outv = (k == 0) ? 0.0f : ((k == 7) ? 6.0f : mix);
    outk = k;
}

__global__ void __launch_bounds__(256)
fp4_soft_ste_kernel(const float* __restrict__ x,
                    const float* __restrict__ delta_raw,
                    float* __restrict__ outAbs,
                    int*   __restrict__ outOrd,
                    uint32_t numChunks)
{
    // Typed LDS: x tiles (16 tiles * 32 float4 = 8 KB) + deltas (64 floats).
    __shared__ v4f   sX[NTILES * CHUNK_V4];
    __shared__ float sD[NTILES * 4u];

    const uint32_t lane  = threadIdx.x & 31u;   // wave32
    const uint32_t wv    = threadIdx.x >> 5;
    const uint32_t gwave = blockIdx.x * WAVES_PB + wv;
    const uint32_t strd  = gridDim.x * WAVES_PB;

    uint32_t c = gwave;
    if (c >= numChunks) return;                 // no barriers used -> safe

    // Branchless issue of one wave-tile's CDNA5 async copies.
    // Exactly 2 ASYNCcnt increments per tile (1x b128 + 1x b32), completing
    // in order, which the s_wait_asynccnt values below rely on.
    auto issue_tile = [&](uint32_t chunk, uint32_t tile) {
        uint32_t xLds = lds_offset(&sX[tile * CHUNK_V4 + lane]);
        uint32_t xOff = (chunk * CHUNK_V4 + lane) * 16u;     // byte offset < 2^31
        asm volatile("global_load_async_to_lds_b128 %0, %1, %2"
                     :: "v"(xLds), "v"(xOff), "s"(x) : "memory");
        // 4 deltas per tile; lanes sharing a block replicate the same dword
        // (same value to same LDS address -> benign, keeps EXEC full).
        uint32_t dLds = lds_offset(&sD[tile * 4u + (lane >> 3)]);
        uint32_t dOff = (chunk * 4u + (lane >> 3)) * 4u;
        asm volatile("global_load_async_to_lds_b32 %0, %1, %2"
                     :: "v"(dLds), "v"(dOff), "s"(delta_raw) : "memory");
    };

    uint32_t buf = 0u;
    issue_tile(c, wv * 2u + buf);               // prologue: fill buffer 0

    for (;;) {
        uint32_t cn = c + strd;
        if (cn < numChunks) {
            // Prefetch next tile into the other buffer, then wait until only
            // those 2 async ops remain outstanding => current tile resident.
            issue_tile(cn, wv * 2u + (buf ^ 1u));
            asm volatile("s_wait_asynccnt 0x2" ::: "memory");
        } else {
            asm volatile("s_wait_asynccnt 0x0" ::: "memory");
        }

        const uint32_t tile = wv * 2u + buf;
        v4f   xv = sX[tile * CHUNK_V4 + lane];          // ds_load_b128
        float dr = sD[tile * 4u + (lane >> 3)];         // ds_load_b32

        // delta = 0.5 * tanh(delta_raw[block]), via exp: (e^2d-1)/(e^2d+1)
        float drc   = fminf(fmaxf(dr, -15.0f), 15.0f);
        float e2    = __expf(2.0f * drc);
        float delta = 0.5f * ((e2 - 1.0f) * __builtin_amdgcn_rcpf(e2 + 1.0f));

        v4f r; v4i kv;
#pragma unroll
        for (int j = 0; j < 4; ++j) {
            float v; int kk;
            quant1(xv[j], delta, v, kk);
            r[j] = v; kv[j] = kk;
        }

        const uint32_t fi = c * CHUNK_V4 + lane;        // float4 index
        __builtin_nontemporal_store(r,  reinterpret_cast<v4f*>(outAbs) + fi);
        __builtin_nontemporal_store(kv, reinterpret_cast<v4i*>(outOrd) + fi);

        if (cn >= numChunks) break;
        c = cn; buf ^= 1u;
    }
}

extern "C" void kernel_launch(void* const* d_in, const int* in_sizes, int n_in,
                              void* d_out, int out_size, void* d_ws, size_t ws_size,
                              hipStream_t stream)
{
    const float* x  = (const float*)d_in[0];     // x_scaled, fp32 [4096*8192]
    const float* dr = (const float*)d_in[1];     // delta_raw, fp32 [n/32]
    const uint32_t n = (uint32_t)in_sizes[0];    // 33,554,432 (divisible by 128)

    float* outAbs = (float*)d_out;               // abs_mix  (fp32, n elems)
    int*   outOrd = (int*)d_out + n;             // ord_hard (i32,  n elems)

    uint32_t numChunks = n / 128u;               // 128 elems per wave-chunk
    uint32_t blocks = (numChunks + WAVES_PB - 1u) / WAVES_PB;
    if (blocks > 2048u) blocks = 2048u;          // 16,384 waves, ~16 chunks each
    if (blocks == 0u)   blocks = 1u;

    fp4_soft_ste_kernel<<<dim3(blocks), dim3(256), 0, stream>>>(
        x, dr, outAbs, outOrd, numChunks);
}